// DenseCapsule_47021301956942
// MI455X (gfx1250) — compile-verified
//
#include <hip/hip_runtime.h>

// Problem constants (match reference setup_inputs)
#define B_SZ   64
#define C_IN   2048
#define D_IN   8
#define K_OUT  32
#define D_OUT  16
#define EPS_SQ 1e-7f

typedef float v2f __attribute__((ext_vector_type(2)));
typedef float v8f __attribute__((ext_vector_type(8)));

// ---- CDNA5 async global->LDS copy (one b128 per lane: 32 lanes * 16B = 512B tile) ----
__device__ __forceinline__ void async_tile_b128(unsigned lds_byte, const float* gaddr) {
  asm volatile("global_load_async_to_lds_b128 %0, %1, off"
               :: "v"(lds_byte), "v"(gaddr) : "memory");
}
__device__ __forceinline__ void wait_async_le1() { asm volatile("s_wait_asynccnt 0x1" ::: "memory"); }
__device__ __forceinline__ void wait_async_le4() { asm volatile("s_wait_asynccnt 0x4" ::: "memory"); }

__device__ __forceinline__ v8f wmma_f32_k4(v2f a, v2f b, v8f c) {
  return __builtin_amdgcn_wmma_f32_16x16x4_f32(false, a, false, b, (short)0, c, false, false);
}

// 16-lane butterfly sum (stays inside lane halves on wave32)
__device__ __forceinline__ float sum16(float p) {
  p += __shfl_xor(p, 1); p += __shfl_xor(p, 2);
  p += __shfl_xor(p, 4); p += __shfl_xor(p, 8);
  return p;
}

// =====================================================================================
// Kernel A: s1[b,k,i] = sum_{c,j} x[b,c,j] W[k,c,i,j]; v1 = squash(s1/32)
// grid (32 k, 4 btile) x 256 threads (8 waves, each wave reduces 256 c)
// =====================================================================================
__global__ __launch_bounds__(256)
void caps_v1_kernel(const float* __restrict__ x, const float* __restrict__ W,
                    float* __restrict__ v1)
{
  const int k    = blockIdx.x;
  const int bt   = blockIdx.y;
  const int tid  = threadIdx.x;
  const int wv   = tid >> 5;
  const int lane = tid & 31;
  const int hl   = lane >> 4;   // lane half
  const int l16  = lane & 15;

  __shared__ float wstage[8][2][128];   // per-wave double-buffered W tile (i-major [16][8])
  __shared__ float red[8][256];
  __shared__ float s_lds[256];

  // A fragment: lane -> row m=l16; VGPR0/1 hold K = h, h+1 with h = 2*half
  const int hA = hl * 2;
  const float* xrow = x + (size_t)(bt * 16 + l16) * C_IN * D_IN;
  // B fragment: lane -> col n=l16; VGPR0/1 hold rows r0, r0+1 with r0 = 2*half
  const int n  = l16;
  const int r0 = hl * 2;

  const unsigned st0 = (unsigned)(uintptr_t)(&wstage[wv][0][0]);
  const unsigned st1 = (unsigned)(uintptr_t)(&wstage[wv][1][0]);
  const float* wk = W + (size_t)k * C_IN * (D_OUT * D_IN);

  const int c0 = wv * (C_IN / 8), c1 = c0 + (C_IN / 8);

  // prologue: stage W tile for c0 (buffer parity c&1; c0 is even)
  async_tile_b128(st0 + (unsigned)(lane * 16), wk + (size_t)c0 * 128 + lane * 4);

  v8f acc0 = {}, acc1 = {};
  for (int c = c0; c < c1; ++c) {
    const int cn = (c + 1 < c1) ? c + 1 : c;                 // clamp (duplicate issue is harmless)
    const unsigned nst = ((c + 1) & 1) ? st1 : st0;
    async_tile_b128(nst + (unsigned)(lane * 16), wk + (size_t)cn * 128 + lane * 4);
    __builtin_prefetch(xrow + (size_t)(c + 8) * D_IN, 0, 0); // global_prefetch_b8
    wait_async_le1();                                        // current tile landed

    const float* tile = &wstage[wv][c & 1][0];
    v2f a0 = *(const v2f*)(xrow + (size_t)c * D_IN + hA);
    v2f b0 = *(const v2f*)(tile + n * 8 + r0);
    acc0 = wmma_f32_k4(a0, b0, acc0);
    v2f a1 = *(const v2f*)(xrow + (size_t)c * D_IN + 4 + hA);
    v2f b1 = *(const v2f*)(tile + n * 8 + 4 + r0);
    acc1 = wmma_f32_k4(a1, b1, acc1);
  }

  v8f accs = acc0 + acc1;
#pragma unroll
  for (int r = 0; r < 8; ++r) red[wv][r * 32 + lane] = accs[r];
  __syncthreads();

  // cross-wave reduce; remap C layout (vgpr r, lane l) -> s[m][n], scale by 1/32
  {
    float s = 0.f;
#pragma unroll
    for (int w2 = 0; w2 < 8; ++w2) s += red[w2][tid];
    const int r = tid >> 5, l = tid & 31;
    s_lds[(r + 8 * (l >> 4)) * 16 + (l & 15)] = s * (1.0f / 32.0f);
  }
  __syncthreads();

  // squash per (b,k): each 16-lane half owns one b row
  {
    const int bl = 2 * wv + hl;
    float e  = s_lds[bl * 16 + l16];
    float sq = sum16(e * e);
    float v  = (sq / (1.0f + sq)) * e / sqrtf(sq + EPS_SQ);
    v1[((size_t)(bt * 16 + bl) * K_OUT + k) * D_OUT + l16] = v;
  }
}

// =====================================================================================
// Kernel B: routing iteration 2 (fused): per c -> u tiles (WMMA), agreement (shuffle
// row-dots vs v1), softmax over k (LDS), s2 += cc*u. Partials per c-chunk to ws.
// grid (32 c-chunks of 64, 4 btile) x 256 threads; wave w owns k = 4w..4w+3.
// =====================================================================================
__global__ __launch_bounds__(256)
void caps_route_kernel(const float* __restrict__ x, const float* __restrict__ W,
                       const float* __restrict__ v1, float* __restrict__ part)
{
  const int chunk = blockIdx.x;
  const int bt    = blockIdx.y;
  const int tid   = threadIdx.x;
  const int wv    = tid >> 5;
  const int lane  = tid & 31;
  const int hl    = lane >> 4;
  const int l16   = lane & 15;
  const int kbase = wv * 4;

  __shared__ float wstage[8][2][4][128];  // per-wave double-buffered 4 W tiles (32KB)
  __shared__ float agr[16][32];
  __shared__ float ccs[16][32];

  const int hA = hl * 2;
  const int n  = l16;
  const int r0 = hl * 2;
  const float* xrow = x + (size_t)(bt * 16 + l16) * C_IN * D_IN;

  // preload v1 tiles in C-matrix layout: vgpr r, lane -> v1[b = r+8*half, k, i = l16]
  v8f v1t[4];
#pragma unroll
  for (int kk = 0; kk < 4; ++kk)
#pragma unroll
    for (int r = 0; r < 8; ++r)
      v1t[kk][r] = v1[((size_t)(bt * 16 + r + 8 * hl) * K_OUT + (kbase + kk)) * D_OUT + l16];

  const int c0 = chunk * 64, c1 = c0 + 64;
  v8f s2acc[4];
#pragma unroll
  for (int kk = 0; kk < 4; ++kk) { v8f z = {}; s2acc[kk] = z; }

  // prologue: stage 4 W tiles for c0
#pragma unroll
  for (int kk = 0; kk < 4; ++kk)
    async_tile_b128((unsigned)(uintptr_t)(&wstage[wv][0][kk][0]) + (unsigned)(lane * 16),
                    W + ((size_t)(kbase + kk) * C_IN + c0) * 128 + lane * 4);

  for (int c = c0; c < c1; ++c) {
    const int cn = (c + 1 < c1) ? c + 1 : c;
#pragma unroll
    for (int kk = 0; kk < 4; ++kk)
      async_tile_b128((unsigned)(uintptr_t)(&wstage[wv][(c + 1) & 1][kk][0]) + (unsigned)(lane * 16),
                      W + ((size_t)(kbase + kk) * C_IN + cn) * 128 + lane * 4);
    wait_async_le4();   // in-order ASYNCcnt: <=4 outstanding => current group done

    v2f a0 = *(const v2f*)(xrow + (size_t)c * D_IN + hA);
    v2f a1 = *(const v2f*)(xrow + (size_t)c * D_IN + 4 + hA);

    v8f u[4];
#pragma unroll
    for (int kk = 0; kk < 4; ++kk) {
      const float* tile = &wstage[wv][c & 1][kk][0];
      v2f b0 = *(const v2f*)(tile + n * 8 + r0);
      v2f b1 = *(const v2f*)(tile + n * 8 + 4 + r0);
      v8f acc = {};
      acc = wmma_f32_k4(a0, b0, acc);
      acc = wmma_f32_k4(a1, b1, acc);
      u[kk] = acc;
    }

    // agreement[b,k,c] = <u[b,k,c,:], v1[b,k,:]> : row dot via 16-lane butterflies
#pragma unroll
    for (int kk = 0; kk < 4; ++kk)
#pragma unroll
      for (int r = 0; r < 8; ++r) {
        float p = sum16(u[kk][r] * v1t[kk][r]);
        if (l16 == 0) agr[r + 8 * hl][kbase + kk] = p;
      }
    __syncthreads();

    // softmax over k (axis=1) per b for this c
    if (tid < 16) {
      float m = agr[tid][0];
#pragma unroll
      for (int q = 1; q < 32; ++q) m = fmaxf(m, agr[tid][q]);
      float s = 0.f;
#pragma unroll
      for (int q = 0; q < 32; ++q) { float e = __expf(agr[tid][q] - m); ccs[tid][q] = e; s += e; }
      float inv = 1.0f / s;
#pragma unroll
      for (int q = 0; q < 32; ++q) ccs[tid][q] *= inv;
    }
    __syncthreads();

    // s2[b,k,i] += cc[b,k,c] * u[b,k,c,i]
#pragma unroll
    for (int kk = 0; kk < 4; ++kk)
#pragma unroll
      for (int r = 0; r < 8; ++r)
        s2acc[kk][r] += ccs[r + 8 * hl][kbase + kk] * u[kk][r];
  }

  // deterministic per-chunk partials: part[chunk][b][k][i]
#pragma unroll
  for (int kk = 0; kk < 4; ++kk)
#pragma unroll
    for (int r = 0; r < 8; ++r)
      part[(((size_t)chunk * B_SZ + (bt * 16 + r + 8 * hl)) * K_OUT + (kbase + kk)) * D_OUT + l16]
          = s2acc[kk][r];
}

// =====================================================================================
// Kernel C: reduce 32 chunk-partials, squash -> output v[b,k,i]
// =====================================================================================
__global__ __launch_bounds__(256)
void caps_out_kernel(const float* __restrict__ part, float* __restrict__ out)
{
  const int tid  = threadIdx.x;
  const int wv   = tid >> 5;
  const int lane = tid & 31;
  const int hl   = lane >> 4;
  const int l16  = lane & 15;

  const int g = blockIdx.x * 16 + (2 * wv + hl);   // (b,k) pair 0..2047
  const int b = g >> 5, k = g & 31;

  float e = 0.f;
#pragma unroll
  for (int ch = 0; ch < 32; ++ch)
    e += part[(((size_t)ch * B_SZ + b) * K_OUT + k) * D_OUT + l16];

  float sq = sum16(e * e);
  float v  = (sq / (1.0f + sq)) * e / sqrtf(sq + EPS_SQ);
  out[((size_t)b * K_OUT + k) * D_OUT + l16] = v;
}

extern "C" void kernel_launch(void* const* d_in, const int* in_sizes, int n_in,
                              void* d_out, int out_size, void* d_ws, size_t ws_size,
                              hipStream_t stream) {
  const float* x = (const float*)d_in[0];   // [64,2048,8]
  const float* W = (const float*)d_in[1];   // [32,2048,16,8]
  float* v1   = (float*)d_ws;               // 64*32*16 floats
  float* part = v1 + (size_t)B_SZ * K_OUT * D_OUT;   // 32*64*32*16 floats (~4MB)

  caps_v1_kernel   <<<dim3(K_OUT, 4), 256, 0, stream>>>(x, W, v1);
  caps_route_kernel<<<dim3(32,    4), 256, 0, stream>>>(x, W, v1, part);
  caps_out_kernel  <<<128,            256, 0, stream>>>(part, (float*)d_out);
}